// MultiHeadedAttention_78262894067866
// MI455X (gfx1250) — compile-verified
//
#include <hip/hip_runtime.h>
#include <hip/hip_bf16.h>

typedef float v2f __attribute__((ext_vector_type(2)));
typedef float v4f __attribute__((ext_vector_type(4)));
typedef float v8f __attribute__((ext_vector_type(8)));

#define BB   2
#define NN   128
#define HH   2
#define DMM  512
#define DKK  256
// 1 / sqrt(N * DK) = 1 / sqrt(32768)
#define ATT_SCALE 0.00552427172801990263f

// ---------------------------------------------------------------------------
// Kernel 1: projection GEMM.  Y(rows x ncols) = X(rows x 512) @ W(512 x ncols) + bias
// One wave32 computes one 16x16 tile with V_WMMA_F32_16X16X4_F32 (exact fp32).
// A layout (16x4 f32): lane<16 -> K=0,1 ; lane>=16 -> K=2,3 ; M = lane&15.
// B layout (4x16 f32): mirrored (VGPR0 = K0/K2, VGPR1 = K1/K3, N = lane&15).
// C/D layout: VGPR r -> M = r + (lane<16 ? 0 : 8), N = lane&15.
// ---------------------------------------------------------------------------
__global__ __launch_bounds__(32) void k_proj_gemm(
    const float* __restrict__ X, const float* __restrict__ W,
    const float* __restrict__ bias, float* __restrict__ Y, int ncols)
{
  const int ctiles = ncols >> 4;
  const int rt = blockIdx.x / ctiles;
  const int ct = blockIdx.x % ctiles;
  const int lane = threadIdx.x;                 // 0..31
  const int m = rt * 16 + (lane & 15);
  const int n = ct * 16 + (lane & 15);
  const int koff = (lane < 16) ? 0 : 2;

  const float* __restrict__ xrow = X + m * DMM;
  v8f acc = {};
  for (int k0 = 0; k0 < DMM; k0 += 4) {
    v2f a, b;
    a.x = xrow[k0 + koff];
    a.y = xrow[k0 + koff + 1];
    b.x = W[(k0 + koff)     * ncols + n];
    b.y = W[(k0 + koff + 1) * ncols + n];
    acc = __builtin_amdgcn_wmma_f32_16x16x4_f32(false, a, false, b,
                                                (short)0, acc, false, false);
  }
  const float bv = bias[n];
  const int rbase = rt * 16 + ((lane < 16) ? 0 : 8);
#pragma unroll
  for (int r = 0; r < 8; ++r)
    Y[(rbase + r) * ncols + n] = acc[r] + bv;
}

// ---------------------------------------------------------------------------
// Kernel 2: per-(b,h) scores + softmax via the algebraic decomposition:
//   scores[n,m] = scale*(T - A_n - A_m + C_n + D_m)   (n != m)
//   scores[n,n] = scale*(T - A_n + E_n)
// One block of 128 threads per (b,h); thread n owns attn row n.
// ---------------------------------------------------------------------------
__global__ __launch_bounds__(128) void k_attn(
    const float* __restrict__ Q, const float* __restrict__ K,
    const float* __restrict__ BX, float* __restrict__ att_ws,
    float* __restrict__ att_out)
{
  const int bh = blockIdx.x;            // 0..3
  const int b = bh >> 1, h = bh & 1;
  const int n = threadIdx.x;            // 0..127

  __shared__ float sA[NN], sC[NN], sD[NN], sE[NN], red[NN];

  const float* __restrict__ qrow  = Q  + (b * NN + n) * DMM + h * DKK;
  const float* __restrict__ krow  = K  + (b * NN + n) * DMM + h * DKK;
  const float* __restrict__ bxrow = BX + (b * NN + n) * DKK;

  float A = 0.f, C = 0.f, D = 0.f, E = 0.f;
  for (int d = 0; d < DKK; ++d) {
    const float qv = qrow[d], kv = krow[d], xv = bxrow[d];
    A = fmaf(qv, kv, A);
    C = fmaf(xv, kv, C);
    D = fmaf(qv, xv, D);
    E = fmaf(xv, xv, E);
  }
  sA[n] = A; sC[n] = C; sD[n] = D; sE[n] = E; red[n] = A;
  __syncthreads();
  for (int s = 64; s > 0; s >>= 1) {
    if (n < s) red[n] += red[n + s];
    __syncthreads();
  }
  const float T = red[0];

  // pass 1: row max
  float mx = -1e30f;
  for (int m = 0; m < NN; ++m) {
    const float raw = (m == n) ? (T - A + sE[n])
                               : (T - A - sA[m] + C + sD[m]);
    mx = fmaxf(mx, raw * ATT_SCALE);
  }
  // pass 2: sum of exp
  float sum = 0.f;
  for (int m = 0; m < NN; ++m) {
    const float raw = (m == n) ? (T - A + sE[n])
                               : (T - A - sA[m] + C + sD[m]);
    sum += __expf(raw * ATT_SCALE - mx);
  }
  const float inv = 1.0f / sum;
  // pass 3: write normalized row (workspace + output tuple element 2)
  float* __restrict__ wrow = att_ws  + ((b * HH + h) * NN + n) * NN;
  float* __restrict__ orow = att_out + ((b * HH + h) * NN + n) * NN;
  for (int m = 0; m < NN; ++m) {
    const float raw = (m == n) ? (T - A + sE[n])
                               : (T - A - sA[m] + C + sD[m]);
    const float a = __expf(raw * ATT_SCALE - mx) * inv;
    wrow[m] = a;
    orow[m] = a;
  }
}

// ---------------------------------------------------------------------------
// Kernel 3: n-independent output precomputation (per b):
//   Base = V' @ Wo + bo            (V'[j',c] = v[b, h=j'/64, j=2*(j'%64)+(c/256), c%256])
//   GL   = (bx - v)|even-j @ Wo[0:256,:]
//   GR   = (bx - v)|odd-j  @ Wo[256:512,:]
// blockIdx = (tile, which, b); one wave32 per 16x16 tile; fp32 WMMA.
// ---------------------------------------------------------------------------
__global__ __launch_bounds__(32) void k_post_gemm(
    const float* __restrict__ V, const float* __restrict__ BX,
    const float* __restrict__ Wo, const float* __restrict__ bo,
    float* __restrict__ Base, float* __restrict__ GLp, float* __restrict__ GRp)
{
  const int which = blockIdx.y;          // 0=Base, 1=GL, 2=GR (uniform)
  const int b = blockIdx.z;
  const int rt = blockIdx.x >> 5;        // 8 row tiles (128 rows)
  const int ct = blockIdx.x & 31;        // 32 col tiles (512 cols)
  const int lane = threadIdx.x;
  const int m = rt * 16 + (lane & 15);   // j' row of A-matrix
  const int n = ct * 16 + (lane & 15);   // output column
  const int koff = (lane < 16) ? 0 : 2;

  const int h = m >> 6;
  const int jhalf = m & 63;
  const int Ksz = (which == 0) ? DMM : DKK;
  const float* __restrict__ Wbase = (which == 2) ? (Wo + DKK * DMM) : Wo;

  v8f acc = {};
  for (int k0 = 0; k0 < Ksz; k0 += 4) {
    const int c0 = k0 + koff;            // even, c0 and c0+1 share j
    float a0, a1;
    if (which == 0) {
      const int j  = 2 * jhalf + (c0 >> 8);
      const int dk = c0 & 255;
      const float* vp = V + (b * NN + j) * DMM + h * DKK + dk;
      a0 = vp[0];
      a1 = vp[1];
    } else {
      const int j = 2 * jhalf + (which - 1);
      const float* vp = V  + (b * NN + j) * DMM + h * DKK + c0;
      const float* xp = BX + (b * NN + j) * DKK + c0;
      a0 = xp[0] - vp[0];
      a1 = xp[1] - vp[1];
    }
    v2f a, bb;
    a.x = a0; a.y = a1;
    bb.x = Wbase[(c0)     * DMM + n];
    bb.y = Wbase[(c0 + 1) * DMM + n];
    acc = __builtin_amdgcn_wmma_f32_16x16x4_f32(false, a, false, bb,
                                                (short)0, acc, false, false);
  }

  float* __restrict__ Y = (which == 0) ? Base : ((which == 1) ? GLp : GRp);
  const float add = (which == 0) ? bo[n] : 0.0f;
  const int rbase = rt * 16 + ((lane < 16) ? 0 : 8);
#pragma unroll
  for (int r = 0; r < 8; ++r)
    Y[(b * NN + rbase + r) * DMM + n] = acc[r] + add;
}

// ---------------------------------------------------------------------------
// Kernel 4: streaming combine — the bandwidth floor (67 MB fp32 output).
//   out[b,n,j',:] = Base[b,j'] + aL*GL[b,j'] + aR*GR[b,j']
// Base/GL/GR (1.5 MB total) live in L2; output stored non-temporally
// (clang ext-vector v4f, required by __builtin_nontemporal_store).
// ---------------------------------------------------------------------------
__global__ __launch_bounds__(256) void k_combine(
    const float* __restrict__ Base, const float* __restrict__ GLp,
    const float* __restrict__ GRp, const float* __restrict__ att,
    float* __restrict__ out)
{
  const long idx4 = (long)blockIdx.x * blockDim.x + threadIdx.x; // v4f index
  const int co4 = (int)(idx4 & 127);
  long t = idx4 >> 7;
  const int jp = (int)(t & 127); t >>= 7;
  const int n  = (int)(t & 127); t >>= 7;
  const int b  = (int)t;

  const int h = jp >> 6;
  const int j = 2 * (jp & 63);
  const float* __restrict__ arow = att + ((b * HH + h) * NN + n) * NN;
  const float aL = arow[j];
  const float aR = arow[j + 1];

  const v4f bb = ((const v4f*)(Base + (b * NN + jp) * DMM))[co4];
  const v4f ll = ((const v4f*)(GLp  + (b * NN + jp) * DMM))[co4];
  const v4f rr = ((const v4f*)(GRp  + (b * NN + jp) * DMM))[co4];

  v4f o;
  o.x = fmaf(aR, rr.x, fmaf(aL, ll.x, bb.x));
  o.y = fmaf(aR, rr.y, fmaf(aL, ll.y, bb.y));
  o.z = fmaf(aR, rr.z, fmaf(aL, ll.z, bb.z));
  o.w = fmaf(aR, rr.w, fmaf(aL, ll.w, bb.w));
  __builtin_nontemporal_store(o, ((v4f*)out) + idx4);
}

// ---------------------------------------------------------------------------
extern "C" void kernel_launch(void* const* d_in, const int* in_sizes, int n_in,
                              void* d_out, int out_size, void* d_ws, size_t ws_size,
                              hipStream_t stream) {
  (void)in_sizes; (void)n_in; (void)out_size; (void)ws_size;
  const float* query = (const float*)d_in[0];
  const float* key   = (const float*)d_in[1];
  const float* value = (const float*)d_in[2];
  const float* boxes = (const float*)d_in[3];
  const float* Wq = (const float*)d_in[4];  const float* bq = (const float*)d_in[5];
  const float* Wk = (const float*)d_in[6];  const float* bk = (const float*)d_in[7];
  const float* Wv = (const float*)d_in[8];  const float* bv = (const float*)d_in[9];
  const float* Wbox = (const float*)d_in[10]; const float* bbox = (const float*)d_in[11];
  const float* Wo = (const float*)d_in[12]; const float* bo = (const float*)d_in[13];

  float* ws = (float*)d_ws;
  float* Q    = ws;                   // 256*512
  float* Kp   = Q    + 131072;        // 256*512
  float* Vp   = Kp   + 131072;        // 256*512
  float* BX   = Vp   + 131072;        // 256*256
  float* ATT  = BX   + 65536;         // 2*2*128*128
  float* Base = ATT  + 65536;         // 2*128*512
  float* GL   = Base + 131072;        // 2*128*512
  float* GR   = GL   + 131072;        // 2*128*512

  float* out_x   = (float*)d_out;                 // 2*128*128*512
  float* out_att = out_x + (long)BB * NN * NN * DMM;

  // 1) projections (fp32 WMMA): 16 row-tiles x {32,16} col-tiles
  k_proj_gemm<<<16 * 32, 32, 0, stream>>>(query, Wq, bq, Q,  DMM);
  k_proj_gemm<<<16 * 32, 32, 0, stream>>>(key,   Wk, bk, Kp, DMM);
  k_proj_gemm<<<16 * 32, 32, 0, stream>>>(value, Wv, bv, Vp, DMM);
  k_proj_gemm<<<16 * 16, 32, 0, stream>>>(boxes, Wbox, bbox, BX, DKK);

  // 2) scores + softmax (one block per (b,h))
  k_attn<<<BB * HH, NN, 0, stream>>>(Q, Kp, BX, ATT, out_att);

  // 3) Base / GL / GR precompute (fp32 WMMA)
  k_post_gemm<<<dim3(8 * 32, 3, BB), 32, 0, stream>>>(Vp, BX, Wo, bo, Base, GL, GR);

  // 4) streaming combine into the 67 MB output
  k_combine<<<(BB * NN * NN * DMM / 4) / 256, 256, 0, stream>>>(Base, GL, GR, ATT, out_x);
}